// PaDiM2_demo_16269336117885
// MI455X (gfx1250) — compile-verified
//
#include <hip/hip_runtime.h>
#include <hip/hip_bf16.h>

typedef float v2f __attribute__((ext_vector_type(2)));
typedef float v4f __attribute__((ext_vector_type(4)));
typedef float v8f __attribute__((ext_vector_type(8)));

#define N_    8
#define CT    448
#define C_    100
#define CPAD  104          // 13 chunks of 8
#define H_    56
#define W_    56
#define PIX   (H_*W_)      // 3136
#define PB    16           // pixels per block (16 floats = 64B segments)
#define NTIL  7            // ceil(100/16) d-tiles
#define OH    224
#define OW    224

// ---------------------------------------------------------------------------
// Main kernel: one workgroup = (16-pixel block, one 16-wide d-tile).
// 256 threads = 8 waves; wave w handles pixels {2w, 2w+1}.
// Symmetry: s = 2 * x^T U x with U = strict-upper(cov) + 0.5*diag(cov),
// so d-tile t only consumes c-chunks 0 .. min(2(t+1),13)-1.
// Left' = X * U_tile via V_WMMA_F32_16X16X4_F32; deterministic (no atomics).
// ---------------------------------------------------------------------------
__global__ __launch_bounds__(256) void padim_mahal_kernel(
    const float* __restrict__ fmaps, const int* __restrict__ sel,
    const float* __restrict__ mean,  const float* __restrict__ cov,
    float* __restrict__ part)
{
    // X staged as [c(104)][p(16)][n(8)] (+8 pad so unconditional m<16 reads stay in-array)
    __shared__ float Xs[CPAD * PB * 8 + 8];   // 53,280 B
    // Cov chunk staged as [c(8)][d(16)][p(17 padded)] -> stride-17 kills bank conflicts
    __shared__ float Cs[8 * 16 * 17];         //  8,704 B

    const int pixbase = blockIdx.x * PB;      // 196 blocks
    const int t       = blockIdx.y;           // d-tile 0..6
    const int tid     = threadIdx.x;
    const int nch     = (2 * (t + 1) < 13) ? 2 * (t + 1) : 13;  // c-chunks needed

    // ---- stage X = fmaps[:,sel,:]-mean for this pixel block, c < nch*8 only ----
    for (int k = 0; k < nch; ++k) {           // k covers 8 c-rows (256 v4f elems)
        int i   = tid + k * 256;
        int c   = i >> 5;                     // 0 .. nch*8-1
        int rem = i & 31;
        int n   = rem >> 2;                   // 0..7
        int p4  = rem & 3;                    // v4f index over 16 pixels
        v4f v = {0.f, 0.f, 0.f, 0.f};
        if (c < C_) {
            int ch = sel[c];
            const v4f* fp = reinterpret_cast<const v4f*>(
                fmaps + (size_t)(n * CT + ch) * PIX + pixbase);
            const v4f* mp = reinterpret_cast<const v4f*>(
                mean + (size_t)c * PIX + pixbase);
            v = fp[p4] - mp[p4];
        }
        int p = p4 * 4;
        float* dst = &Xs[(c * PB + p) * 8 + n];
        dst[0] = v[0]; dst[8] = v[1]; dst[16] = v[2]; dst[24] = v[3];
    }
    __syncthreads();

    const int lane = tid & 31;
    const int wv   = tid >> 5;                // 0..7
    const int m    = lane & 15;               // A/B row-within-half
    const int hi   = lane >> 4;               // 0: K+0/K+1, 1: K+2/K+3
    const int p0   = wv * 2;
    const int p1   = wv * 2 + 1;

    v8f acc0 = {0.f,0.f,0.f,0.f,0.f,0.f,0.f,0.f};
    v8f acc1 = {0.f,0.f,0.f,0.f,0.f,0.f,0.f,0.f};

    // ---- K loop over upper-triangular c-chunks ----
    for (int chv = 0; chv < nch; ++chv) {
        const int c0 = chv * 8;

        // stage U[c0..c0+7][t*16..+15][16 pixels] : up to 512 v4f NT loads
        for (int k = 0; k < 2; ++k) {
            int i   = tid + k * 256;
            int row = i >> 2;                 // 0..127 = cl*16+dl
            int p4  = i & 3;
            int cl  = row >> 4;
            int dl  = row & 15;
            int cg  = c0 + cl;
            int dg  = t * 16 + dl;
            v4f v = {0.f, 0.f, 0.f, 0.f};
            float w = 0.f;
            if (cg < C_ && dg < C_)
                w = (cg < dg) ? 1.f : ((cg == dg) ? 0.5f : 0.f);
            if (w != 0.f) {
                const v4f* cp = reinterpret_cast<const v4f*>(
                    cov + (size_t)(cg * C_ + dg) * PIX + pixbase);
                v = __builtin_nontemporal_load(cp + p4) * w;   // NT: don't pollute L2
            }
            float* dst = &Cs[(cl * 16 + dl) * 17 + p4 * 4];
            dst[0] = v[0]; dst[1] = v[1]; dst[2] = v[2]; dst[3] = v[3];
        }
        __syncthreads();

        // two K=4 WMMA steps per chunk, per pixel
#pragma unroll
        for (int kk = 0; kk < 2; ++kk) {
            const int cl = kk * 4 + hi * 2;   // local c for this lane-half
            // pixel 0
            {
                float a0 = Xs[((c0 + cl    ) * PB + p0) * 8 + m];
                float a1 = Xs[((c0 + cl + 1) * PB + p0) * 8 + m];
                v2f A, B;
                A[0] = (m < 8) ? a0 : 0.f;    // v_cndmask, no exec juggling
                A[1] = (m < 8) ? a1 : 0.f;
                B[0] = Cs[((cl    ) * 16 + m) * 17 + p0];
                B[1] = Cs[((cl + 1) * 16 + m) * 17 + p0];
                acc0 = __builtin_amdgcn_wmma_f32_16x16x4_f32(
                           false, A, false, B, (short)0, acc0, false, false);
            }
            // pixel 1
            {
                float a0 = Xs[((c0 + cl    ) * PB + p1) * 8 + m];
                float a1 = Xs[((c0 + cl + 1) * PB + p1) * 8 + m];
                v2f A, B;
                A[0] = (m < 8) ? a0 : 0.f;
                A[1] = (m < 8) ? a1 : 0.f;
                B[0] = Cs[((cl    ) * 16 + m) * 17 + p1];
                B[1] = Cs[((cl + 1) * 16 + m) * 17 + p1];
                acc1 = __builtin_amdgcn_wmma_f32_16x16x4_f32(
                           false, A, false, B, (short)0, acc1, false, false);
            }
        }
        __syncthreads();
    }

    // ---- partial dot: s_part[n] = sum_{d in tile} Left'[n][d] * X[n][d] ----
    // Lane holds Left'[n=r (+8 if hi)][d = t*16 + m] in acc[r]; hi rows are zero.
    int dg  = t * 16 + m;
    int dcl = (dg < CPAD) ? dg : (CPAD - 1); // clamped; acc cols for d>=100 are 0
#pragma unroll
    for (int r = 0; r < 8; ++r) {
        float x0 = Xs[(dcl * PB + p0) * 8 + r];
        float x1 = Xs[(dcl * PB + p1) * 8 + r];
        float v0 = (hi == 0) ? acc0[r] * x0 : 0.f;
        float v1 = (hi == 0) ? acc1[r] * x1 : 0.f;
        v0 += __shfl_xor(v0, 1, 16); v0 += __shfl_xor(v0, 2, 16);
        v0 += __shfl_xor(v0, 4, 16); v0 += __shfl_xor(v0, 8, 16);
        v1 += __shfl_xor(v1, 1, 16); v1 += __shfl_xor(v1, 2, 16);
        v1 += __shfl_xor(v1, 4, 16); v1 += __shfl_xor(v1, 8, 16);
        if (lane == 0) {
            part[(size_t)(t * 8 + r) * PIX + pixbase + p0] = v0;
            part[(size_t)(t * 8 + r) * PIX + pixbase + p1] = v1;
        }
    }
}

// ---------------------------------------------------------------------------
// Combine the 7 d-tile partials, apply symmetry factor 2, sqrt -> s_map
// ---------------------------------------------------------------------------
__global__ __launch_bounds__(256) void padim_combine_kernel(
    const float* __restrict__ part, float* __restrict__ smap)
{
    int i = blockIdx.x * 256 + threadIdx.x;
    if (i >= N_ * PIX) return;
    int n = i / PIX, pix = i % PIX;
    float s = 0.f;
#pragma unroll
    for (int t = 0; t < NTIL; ++t)
        s += part[(size_t)(t * 8 + n) * PIX + pix];
    smap[i] = sqrtf(fmaxf(2.f * s, 0.f));    // x^T C x = 2 * x^T U x
}

// ---------------------------------------------------------------------------
// x4 bilinear upsample (half-pixel, edge clamp) + normalize
// ---------------------------------------------------------------------------
__global__ __launch_bounds__(256) void padim_upsample_kernel(
    const float* __restrict__ smap,
    const float* __restrict__ mn_p, const float* __restrict__ mx_p,
    float* __restrict__ out)
{
    int i = blockIdx.x * 256 + threadIdx.x;
    if (i >= N_ * OH * OW) return;
    int n   = i / (OH * OW);
    int rem = i % (OH * OW);
    int oy  = rem / OW, ox = rem % OW;
    float fy = 0.25f * oy - 0.375f;           // (o+0.5)/4 - 0.5
    float fx = 0.25f * ox - 0.375f;
    int iy0 = (int)floorf(fy); float wy = fy - (float)iy0;
    int ix0 = (int)floorf(fx); float wx = fx - (float)ix0;
    int y0 = min(max(iy0, 0), H_ - 1), y1 = min(max(iy0 + 1, 0), H_ - 1);
    int x0 = min(max(ix0, 0), W_ - 1), x1 = min(max(ix0 + 1, 0), W_ - 1);
    const float* sp = smap + (size_t)n * PIX;
    float v00 = sp[y0 * W_ + x0], v01 = sp[y0 * W_ + x1];
    float v10 = sp[y1 * W_ + x0], v11 = sp[y1 * W_ + x1];
    float v = (1.f - wy) * ((1.f - wx) * v00 + wx * v01)
            +         wy * ((1.f - wx) * v10 + wx * v11);
    float mn = *mn_p, mx = *mx_p;
    out[i] = (v - mn) / (mx - mn);
}

// ---------------------------------------------------------------------------
extern "C" void kernel_launch(void* const* d_in, const int* in_sizes, int n_in,
                              void* d_out, int out_size, void* d_ws, size_t ws_size,
                              hipStream_t stream) {
    const float* fmaps = (const float*)d_in[0];
    const int*   sel   = (const int*)  d_in[1];
    const float* mean  = (const float*)d_in[2];
    const float* cov   = (const float*)d_in[3];
    const float* mn    = (const float*)d_in[4];
    const float* mx    = (const float*)d_in[5];
    float* out  = (float*)d_out;

    float* part = (float*)d_ws;                    // 7*8*3136 floats = 702,464 B
    float* smap = part + (size_t)NTIL * N_ * PIX;  // 8*3136 floats

    dim3 grid2(PIX / PB, NTIL);                    // 196 x 7 = 1372 workgroups
    padim_mahal_kernel<<<grid2, 256, 0, stream>>>(fmaps, sel, mean, cov, part);

    padim_combine_kernel<<<(N_ * PIX + 255) / 256, 256, 0, stream>>>(part, smap);

    padim_upsample_kernel<<<(N_ * OH * OW + 255) / 256, 256, 0, stream>>>(
        smap, mn, mx, out);
}